// AdaptiveAnchorGAT_82806969466906
// MI455X (gfx1250) — compile-verified
//
#include <hip/hip_runtime.h>
#include <hip/hip_bf16.h>
#include <math.h>

// ---------------------------------------------------------------------------
// Problem constants (from reference)
// ---------------------------------------------------------------------------
#define B_   4096
#define D_   128
#define NA_  128
#define AD_  128
#define NI_  50000

typedef float v2f __attribute__((ext_vector_type(2)));
typedef float v8f __attribute__((ext_vector_type(8)));

// ---------------------------------------------------------------------------
// Stage A: u = gather(user_table); nf = LN(u); t = nf @ send_W; s2 = t . a2
// One block per row, 128 threads. Small (134 MFLOP total) -> simple VALU.
// ---------------------------------------------------------------------------
__global__ __launch_bounds__(128) void stage_a_ln_gemm(
    const int* __restrict__ uidx, const float* __restrict__ utab,
    const float* __restrict__ g1, const float* __restrict__ b1,
    const float* __restrict__ sendW, const float* __restrict__ send_a,
    float* __restrict__ t_out, float* __restrict__ s2_out)
{
    __shared__ float red[128];
    __shared__ float nf[128];
    const int row = blockIdx.x;
    const int k   = threadIdx.x;
    const float u = utab[(size_t)uidx[row] * D_ + k];

    red[k] = u; __syncthreads();
#pragma unroll
    for (int s = 64; s > 0; s >>= 1) { if (k < s) red[k] += red[k + s]; __syncthreads(); }
    const float mean = red[0] * (1.0f / 128.0f);
    __syncthreads();
    const float d = u - mean;
    red[k] = d * d; __syncthreads();
#pragma unroll
    for (int s = 64; s > 0; s >>= 1) { if (k < s) red[k] += red[k + s]; __syncthreads(); }
    const float var = red[0] * (1.0f / 128.0f);
    __syncthreads();
    nf[k] = d * rsqrtf(var + 1e-5f) * g1[k] + b1[k];
    __syncthreads();

    float acc = 0.0f;
#pragma unroll 8
    for (int kk = 0; kk < 128; ++kk) acc = fmaf(nf[kk], sendW[kk * AD_ + k], acc);
    t_out[(size_t)row * AD_ + k] = acc;

    // s2[row] = t[row,:] . send_a[AD:2AD]   (the a1 half cancels in softmax)
    red[k] = acc * send_a[AD_ + k]; __syncthreads();
#pragma unroll
    for (int s = 64; s > 0; s >>= 1) { if (k < s) red[k] += red[k + s]; __syncthreads(); }
    if (k == 0) s2_out[row] = red[0];
}

// ---------------------------------------------------------------------------
// Stage B1: softmax stats over s2[0..B) (single block, deterministic)
// ---------------------------------------------------------------------------
__global__ __launch_bounds__(1024) void stage_b1_softmax_stats(
    const float* __restrict__ s2, float* __restrict__ stats)
{
    __shared__ float red[1024];
    const int t = threadIdx.x;
    float m = -3.4e38f;
    for (int j = t; j < B_; j += 1024) m = fmaxf(m, s2[j]);
    red[t] = m; __syncthreads();
#pragma unroll
    for (int s = 512; s > 0; s >>= 1) { if (t < s) red[t] = fmaxf(red[t], red[t + s]); __syncthreads(); }
    const float mx = red[0]; __syncthreads();
    float sum = 0.0f;
    for (int j = t; j < B_; j += 1024) sum += __expf(s2[j] - mx);
    red[t] = sum; __syncthreads();
#pragma unroll
    for (int s = 512; s > 0; s >>= 1) { if (t < s) red[t] += red[t + s]; __syncthreads(); }
    if (t == 0) { stats[0] = mx; stats[1] = red[0]; }
}

// ---------------------------------------------------------------------------
// Stage B2: partial weighted sums  part[b][k] = sum_{j in chunk b} p[j]*t[j][k]
// 128 blocks x 128 threads; chunk = 32 rows. Deterministic (no atomics).
// ---------------------------------------------------------------------------
__global__ __launch_bounds__(128) void stage_b2_weighted_partial(
    const float* __restrict__ t_ws, const float* __restrict__ s2,
    const float* __restrict__ stats, float* __restrict__ part)
{
    const int b = blockIdx.x, k = threadIdx.x;
    const float mx = stats[0], inv = 1.0f / stats[1];
    float acc = 0.0f;
    const int j0 = b * (B_ / 128);
#pragma unroll 4
    for (int j = j0; j < j0 + (B_ / 128); ++j)
        acc = fmaf(__expf(s2[j] - mx) * inv, t_ws[(size_t)j * AD_ + k], acc);
    part[(size_t)b * AD_ + k] = acc;
}

// ---------------------------------------------------------------------------
// Stage B3: sent_row -> ap = sent @ anchors^T -> LN -> rec = na @ recv_W ->
//           sinr = sin(rec). Single block, 128 threads. recv_a drops out
//           (uniform softmax over identical rows).
// ---------------------------------------------------------------------------
__global__ __launch_bounds__(128) void stage_b3_anchor_pipeline(
    const float* __restrict__ part, const float* __restrict__ anchors,
    const float* __restrict__ g2, const float* __restrict__ b2,
    const float* __restrict__ recvW, float* __restrict__ sinr)
{
    __shared__ float sent[128];
    __shared__ float na[128];
    __shared__ float red[128];
    const int t = threadIdx.x;

    float s = 0.0f;
#pragma unroll 8
    for (int b = 0; b < 128; ++b) s += part[(size_t)b * AD_ + t];
    sent[t] = s; __syncthreads();

    float ap = 0.0f;
#pragma unroll 8
    for (int k = 0; k < AD_; ++k) ap = fmaf(sent[k], anchors[(size_t)t * AD_ + k], ap);

    red[t] = ap; __syncthreads();
#pragma unroll
    for (int s2_ = 64; s2_ > 0; s2_ >>= 1) { if (t < s2_) red[t] += red[t + s2_]; __syncthreads(); }
    const float mean = red[0] * (1.0f / 128.0f);
    __syncthreads();
    const float d = ap - mean;
    red[t] = d * d; __syncthreads();
#pragma unroll
    for (int s2_ = 64; s2_ > 0; s2_ >>= 1) { if (t < s2_) red[t] += red[t + s2_]; __syncthreads(); }
    const float var = red[0] * (1.0f / 128.0f);
    __syncthreads();
    na[t] = d * rsqrtf(var + 1e-5f) * g2[t] + b2[t];
    __syncthreads();

    float rec = 0.0f;
#pragma unroll 8
    for (int n = 0; n < NA_; ++n) rec = fmaf(na[n], recvW[(size_t)n * D_ + t], rec);
    sinr[t] = sinf(rec);
}

// ---------------------------------------------------------------------------
// Stage D: ue = u + sinr (broadcast), pos = gather(item_table)
// ---------------------------------------------------------------------------
__global__ __launch_bounds__(256) void stage_d_ue_pos(
    const int* __restrict__ uidx, const int* __restrict__ pidx,
    const float* __restrict__ utab, const float* __restrict__ itab,
    const float* __restrict__ sinr,
    float* __restrict__ ue_out, float* __restrict__ pos_out)
{
    const int idx = blockIdx.x * 256 + threadIdx.x;   // < B*D
    const int i = idx >> 7, k = idx & 127;
    ue_out[idx]  = utab[(size_t)uidx[i] * D_ + k] + sinr[k];
    pos_out[idx] = itab[(size_t)pidx[i] * D_ + k];
}

// ---------------------------------------------------------------------------
// Stage C: preds = ue @ item_table^T   (4096 x 50000, K=128, fp32 WMMA)
// Block tile 64(M) x 128(N), whole K staged in LDS. 8 waves: 4 in M x 2 in N,
// each wave owns 16x64 via four 16x16 f32 accumulators, K swept with
// v_wmma_f32_16x16x4_f32 (exact fp32; store-bandwidth bound anyway).
// ---------------------------------------------------------------------------
#define TM 64
#define TN 128
#define KK 128
#define LDP 132   // padded row stride (floats); 132*4B = 33*16B -> float4 aligned

__global__ __launch_bounds__(256) void preds_gemm_wmma(
    const float* __restrict__ ue,    // [B][K]
    const float* __restrict__ item,  // [NI][K]
    float* __restrict__ preds,       // [B][NI]
    int ni)
{
    __shared__ float As[TM][LDP];
    __shared__ float Bs[TN][LDP];

    const int m0  = blockIdx.y * TM;
    const int n0  = blockIdx.x * TN;
    const int tid = threadIdx.x;

    // ---- fill A tile: 64x128 floats = 2048 float4, 8 per thread
#pragma unroll
    for (int v = tid; v < (TM * KK) / 4; v += 256) {
        const int r = v >> 5;            // /(KK/4)
        const int c = (v & 31) << 2;
        const float4 x = *(const float4*)(ue + (size_t)(m0 + r) * KK + c);
        *(float4*)&As[r][c] = x;
    }
    // ---- fill B tile: 128x128 floats = 4096 float4, 16 per thread (guard N)
#pragma unroll
    for (int v = tid; v < (TN * KK) / 4; v += 256) {
        const int r = v >> 5;
        const int c = (v & 31) << 2;
        const int n = n0 + r;
        float4 x = make_float4(0.f, 0.f, 0.f, 0.f);
        if (n < ni) x = *(const float4*)(item + (size_t)n * KK + c);
        *(float4*)&Bs[r][c] = x;
    }
    __syncthreads();

    const int wave = tid >> 5;
    const int lane = tid & 31;
    const int mw = (wave & 3) * 16;   // wave row offset in tile
    const int nw = (wave >> 2) * 64;  // wave col offset in tile
    const int l15   = lane & 15;
    const int khalf = (lane >> 4) * 2;   // lanes 16-31 carry K+2,K+3 (ISA layout)

    const float* Ar  = &As[mw + l15][0];
    const float* Br0 = &Bs[nw +  0 + l15][0];
    const float* Br1 = &Bs[nw + 16 + l15][0];
    const float* Br2 = &Bs[nw + 32 + l15][0];
    const float* Br3 = &Bs[nw + 48 + l15][0];

    v8f c0 = {}, c1 = {}, c2 = {}, c3 = {};
#pragma unroll
    for (int k = 0; k < KK; k += 4) {
        const int kk = k + khalf;
        v2f a;  a[0]  = Ar[kk];  a[1]  = Ar[kk + 1];
        v2f b0; b0[0] = Br0[kk]; b0[1] = Br0[kk + 1];
        v2f b1; b1[0] = Br1[kk]; b1[1] = Br1[kk + 1];
        v2f b2; b2[0] = Br2[kk]; b2[1] = Br2[kk + 1];
        v2f b3; b3[0] = Br3[kk]; b3[1] = Br3[kk + 1];
        c0 = __builtin_amdgcn_wmma_f32_16x16x4_f32(false, a, false, b0, (short)0, c0, false, false);
        c1 = __builtin_amdgcn_wmma_f32_16x16x4_f32(false, a, false, b1, (short)0, c1, false, false);
        c2 = __builtin_amdgcn_wmma_f32_16x16x4_f32(false, a, false, b2, (short)0, c2, false, false);
        c3 = __builtin_amdgcn_wmma_f32_16x16x4_f32(false, a, false, b3, (short)0, c3, false, false);
    }

    // ---- store: C/D layout: VGPR r -> M=r (lanes 0-15) / M=8+r (lanes 16-31)
    const int rowAdd = (lane >> 4) * 8;
    const int cb = n0 + nw + l15;
#pragma unroll
    for (int r = 0; r < 8; ++r) {
        const int row = m0 + mw + rowAdd + r;
        float* prow = preds + (size_t)row * (size_t)ni;
        if (cb      < ni) prow[cb     ] = c0[r];
        if (cb + 16 < ni) prow[cb + 16] = c1[r];
        if (cb + 32 < ni) prow[cb + 32] = c2[r];
        if (cb + 48 < ni) prow[cb + 48] = c3[r];
    }
}

// ---------------------------------------------------------------------------
// Launch
// ---------------------------------------------------------------------------
extern "C" void kernel_launch(void* const* d_in, const int* in_sizes, int n_in,
                              void* d_out, int out_size, void* d_ws, size_t ws_size,
                              hipStream_t stream)
{
    const int*   uidx    = (const int*)  d_in[0];
    const int*   pidx    = (const int*)  d_in[1];
    const float* utab    = (const float*)d_in[2];
    const float* itab    = (const float*)d_in[3];
    const float* ln1_g   = (const float*)d_in[4];
    const float* ln1_b   = (const float*)d_in[5];
    const float* sendW   = (const float*)d_in[6];
    const float* send_a  = (const float*)d_in[7];
    const float* anchors = (const float*)d_in[8];
    const float* ln2_g   = (const float*)d_in[9];
    const float* ln2_b   = (const float*)d_in[10];
    const float* recvW   = (const float*)d_in[11];
    // d_in[12] = recv_a: provably unused (uniform softmax over identical rows)

    float* preds   = (float*)d_out;                       // [B][NI]
    float* ue_out  = preds  + (size_t)B_ * NI_;           // [B][D]
    float* pos_out = ue_out + (size_t)B_ * D_;            // [B][D]

    float* ws    = (float*)d_ws;
    float* t_ws  = ws;                        // B*D      = 524288 floats
    float* s2    = t_ws  + (size_t)B_ * D_;   // B
    float* stats = s2    + B_;                // 2
    float* part  = stats + 2;                 // 128*128
    float* sinr  = part  + 128 * 128;         // 128      (total ~2.2 MB)

    stage_a_ln_gemm<<<B_, 128, 0, stream>>>(uidx, utab, ln1_g, ln1_b,
                                            sendW, send_a, t_ws, s2);
    stage_b1_softmax_stats<<<1, 1024, 0, stream>>>(s2, stats);
    stage_b2_weighted_partial<<<128, 128, 0, stream>>>(t_ws, s2, stats, part);
    stage_b3_anchor_pipeline<<<1, 128, 0, stream>>>(part, anchors, ln2_g, ln2_b,
                                                    recvW, sinr);
    stage_d_ue_pos<<<(B_ * D_) / 256, 256, 0, stream>>>(uidx, pidx, utab, itab,
                                                        sinr, ue_out, pos_out);

    dim3 grid((NI_ + TN - 1) / TN, B_ / TM);   // 391 x 64
    preds_gemm_wmma<<<grid, 256, 0, stream>>>(ue_out, itab, preds, NI_);
}